// MatrixTreeMarkovICL_83511344103448
// MI455X (gfx1250) — compile-verified
//
#include <hip/hip_runtime.h>
#include <math.h>

// MI455X / gfx1250, wave32. fp32 end-to-end; matrix op = V_WMMA_F32_16X16X4_F32.

typedef float v2f __attribute__((ext_vector_type(2)));
typedef float v8f __attribute__((ext_vector_type(8)));

#define N_NODES   64
#define Z_FULL    136     // (N_CTX+1)*Z_DIM
#define N_CTX     16
#define K_CLS     75
#define NB        16      // batch elements per workgroup (fills WMMA N dim)
#define ROWSTRIDE 65      // 64 + 1 pad: row-parallel GE is bank-conflict-free
#define MATSTRIDE 4164    // 64*65 + 4 : inter-matrix stride rotates banks (4 mod 64)
#define XSTRIDE   154     // 152 rounded to even (+2) for 8B-aligned v2f LDS loads

// LDS (floats): mats[NB*MATSTRIDE] | X[NB*XSTRIDE] | rhs[NB*64] | att[NB*16] | lab[NB*16] (int)
#define SMEM_FLOATS (NB*MATSTRIDE + NB*XSTRIDE + NB*64 + NB*16)
#define SMEM_BYTES  (SMEM_FLOATS*4 + NB*16*4)

__global__ __launch_bounds__(256)
void mtm_icl_kernel(const float* __restrict__ z,      // (B,17,8)
                    const int*   __restrict__ labels, // (B,16)
                    const float* __restrict__ Kp,     // (64,64,136)
                    const float* __restrict__ Lm,     // (64,64,16)
                    const float* __restrict__ Bm,     // (64,16)
                    const float* __restrict__ base,   // (64,64)
                    float*       __restrict__ out)    // (B,75)
{
  extern __shared__ float smem[];
  float* mats = smem;                        // NB*MATSTRIDE
  float* X    = mats + NB * MATSTRIDE;       // NB*XSTRIDE  (x vector per batch col)
  float* rhs  = X    + NB * XSTRIDE;         // NB*64 (col-sums, then RHS, then p)
  float* att  = rhs  + NB * 64;              // NB*16
  int*   lab  = (int*)(att + NB * 16);       // NB*16

  const int tid  = threadIdx.x;
  const int lane = tid & 31;
  const int wv   = tid >> 5;                 // 8 waves
  const int b0   = blockIdx.x * NB;

  // ---------- Phase A: stage z_flat + labels into LDS ----------
  for (int idx = tid; idx < NB * Z_FULL; idx += 256) {
    int n = idx / Z_FULL, d = idx - n * Z_FULL;
    X[n * XSTRIDE + d] = z[(size_t)(b0 + n) * Z_FULL + d];
  }
  for (int idx = tid; idx < NB * N_CTX; idx += 256) {
    int n = idx >> 4, c = idx & 15;
    int lv = labels[(size_t)(b0 + n) * N_CTX + c];
    lab[n * 16 + c] = lv;
    X[n * XSTRIDE + Z_FULL + c] = (float)lv;   // label part of the 152-dim vector
  }
  __syncthreads();

  // ---------- Phase B: rate_mod GEMM via f32 WMMA + exp epilogue ----------
  // A frag (16x4 f32): lane = m + 16*khi, vgpr v holds K = 2*khi + v
  // B frag (4x16 f32): lane = n + 16*khi, vgpr v holds K = 2*khi + v
  // C/D    (16x16):    lane = n + 16*khi, vgpr v holds M = v + 8*khi
  const int khi = lane >> 4;
  const int ln  = lane & 15;   // m for A-frag, n for B/C-frags
  for (int rt = wv; rt < 256; rt += 8) {       // 256 row-tiles of W (4096 rows)
    const int r = rt * 16 + ln;
    const float* rowK = Kp + (size_t)r * Z_FULL + 2 * khi;
    const float* rowL = Lm + (size_t)r * N_CTX + 2 * khi;
    const float* xrow = X + ln * XSTRIDE + 2 * khi;
    v8f acc = {0.f, 0.f, 0.f, 0.f, 0.f, 0.f, 0.f, 0.f};
#pragma unroll
    for (int d0 = 0; d0 < Z_FULL; d0 += 4) {   // K_params part (34 WMMAs)
      v2f a = *(const v2f*)(rowK + d0);
      v2f b = *(const v2f*)(xrow + d0);
      acc = __builtin_amdgcn_wmma_f32_16x16x4_f32(false, a, false, b,
                                                  (short)0, acc, false, false);
    }
#pragma unroll
    for (int d0 = 0; d0 < N_CTX; d0 += 4) {    // label_modulation part (4 WMMAs)
      v2f a = *(const v2f*)(rowL + d0);
      v2f b = *(const v2f*)(xrow + Z_FULL + d0);
      acc = __builtin_amdgcn_wmma_f32_16x16x4_f32(false, a, false, b,
                                                  (short)0, acc, false, false);
    }
    // epilogue: log_rates = clip(base + acc); rates = exp, zero diagonal
#pragma unroll
    for (int v = 0; v < 8; ++v) {
      int rg = rt * 16 + v + 8 * khi;          // global W row = i*64+j
      int i = rg >> 6, j = rg & 63;
      float lr = base[rg] + acc[v];
      lr = fminf(fmaxf(lr, -15.f), 3.f);
      float rate = (i == j) ? 0.f : __expf(lr);
      mats[ln * MATSTRIDE + i * ROWSTRIDE + j] = rate;
    }
  }
  __syncthreads();

  // ---------- Phase C: column sums -> diagonal; last row = 1; rhs = e63 ----------
  for (int t = tid; t < NB * 64; t += 256) {
    int b = t >> 6, j = t & 63;
    const float* M = mats + b * MATSTRIDE;
    float s = 0.f;
    for (int i = 0; i < 64; ++i) s += M[i * ROWSTRIDE + j];
    rhs[t] = s;
  }
  __syncthreads();
  for (int t = tid; t < NB * 64; t += 256) {
    int b = t >> 6, j = t & 63;
    float* M = mats + b * MATSTRIDE;
    float cs = rhs[t];
    if (j == 63) {
      M[63 * ROWSTRIDE + 63] = 1.f;
      rhs[t] = 1.f;                            // b = e_63
    } else {
      M[j * ROWSTRIDE + j] = -cs;
      M[63 * ROWSTRIDE + j] = 1.f;
      rhs[t] = 0.f;
    }
  }
  __syncthreads();

  // ---------- Phase D+E: per-wave 64x64 solve, softmax, logits ----------
  for (int nsel = 0; nsel < 2; ++nsel) {
    const int n = wv + nsel * 8;
    float* A = mats + n * MATSTRIDE;
    float* x = rhs + n * 64;
    const int r0 = lane, r1 = lane + 32;       // two rows per lane

    // forward elimination (row-parallel; 65-float stride -> distinct banks)
    for (int k = 0; k < 63; ++k) {
      float pinv = 1.f / A[k * ROWSTRIDE + k];
      bool d0a = (r0 > k), d1a = (r1 > k);
      float f0 = d0a ? A[r0 * ROWSTRIDE + k] * pinv : 0.f;
      float f1 = d1a ? A[r1 * ROWSTRIDE + k] * pinv : 0.f;
      for (int j = k + 1; j < 64; ++j) {
        float akj = A[k * ROWSTRIDE + j];
        if (d0a) A[r0 * ROWSTRIDE + j] -= f0 * akj;
        if (d1a) A[r1 * ROWSTRIDE + j] -= f1 * akj;
      }
      float xk = x[k];
      if (d0a) x[r0] -= f0 * xk;
      if (d1a) x[r1] -= f1 * xk;
      __builtin_amdgcn_wave_barrier();
    }

    // back substitution (column-oriented, lane-parallel)
    for (int k = 63; k >= 0; --k) {
      float xk = x[k] / A[k * ROWSTRIDE + k];
      if (r0 < k) x[r0] -= A[r0 * ROWSTRIDE + k] * xk;
      if (r1 < k) x[r1] -= A[r1 * ROWSTRIDE + k] * xk;
      if (lane == 0) x[k] = xk;
      __builtin_amdgcn_wave_barrier();
    }

    // clip + normalize p
    float v0 = fmaxf(x[r0], 0.f), v1 = fmaxf(x[r1], 0.f);
    float s = v0 + v1;
    for (int off = 16; off > 0; off >>= 1) s += __shfl_xor(s, off, 32);
    float inv = 1.f / s;
    x[r0] = v0 * inv;
    x[r1] = v1 * inv;
    __builtin_amdgcn_wave_barrier();

    // q = p @ B_mat ; softmax over 16 (lanes 16..31 mirror 0..15)
    int c = lane & 15;
    float q = 0.f;
    for (int j = 0; j < 64; ++j) q += x[j] * Bm[j * 16 + c];
    float mx = q;
    for (int off = 8; off > 0; off >>= 1) mx = fmaxf(mx, __shfl_xor(mx, off, 32));
    float e = __expf(q - mx);
    float es = e;
    for (int off = 8; off > 0; off >>= 1) es += __shfl_xor(es, off, 32);
    float a = e / es;
    if (lane < 16) att[n * 16 + c] = a;
    __builtin_amdgcn_wave_barrier();

    // logits: scatter attention by label into 75 classes, then log
    float* op = out + (size_t)(b0 + n) * K_CLS;
    for (int k = lane; k < K_CLS; k += 32) {
      float accv = 1e-10f;
      for (int cc = 0; cc < 16; ++cc)
        accv += (lab[n * 16 + cc] == k + 1) ? att[n * 16 + cc] : 0.f;
      op[k] = logf(accv);
    }
  }
}

extern "C" void kernel_launch(void* const* d_in, const int* in_sizes, int n_in,
                              void* d_out, int out_size, void* d_ws, size_t ws_size,
                              hipStream_t stream) {
  const float* z      = (const float*)d_in[0];
  const int*   labels = (const int*)d_in[1];
  const float* Kp     = (const float*)d_in[2];
  const float* Lm     = (const float*)d_in[3];
  const float* Bm     = (const float*)d_in[4];
  const float* base   = (const float*)d_in[5];
  float* out = (float*)d_out;

  const int B = in_sizes[1] / N_CTX;           // 8192
  const int grid = B / NB;                     // 512 workgroups

  (void)hipFuncSetAttribute(reinterpret_cast<const void*>(&mtm_icl_kernel),
                            hipFuncAttributeMaxDynamicSharedMemorySize,
                            SMEM_BYTES);

  mtm_icl_kernel<<<grid, 256, SMEM_BYTES, stream>>>(z, labels, Kp, Lm, Bm, base, out);
}